// LinearKalmanFilter_55654186221583
// MI455X (gfx1250) — compile-verified
//
#include <hip/hip_runtime.h>

#define T_STEPS 2048
#define BATCH   256
#define NX      128
#define NY      64
#define NU      32
#define ND      16
#define ROWS    16   // batch rows per phase-2 block (16 blocks total)

// LDS row strides padded by +4 floats: stride mod 64 banks == 4, so the 16
// lanes of an A-fragment load (consecutive rows) hit 16 distinct banks.
#define PX (NX + 4)  // 132
#define PY (NY + 4)  // 68
#define PU (NU + 4)  // 36
#define PD (ND + 4)  // 20

typedef float v2f __attribute__((ext_vector_type(2)));
typedef float v8f __attribute__((ext_vector_type(8)));

enum { CI_ZERO = 0, CI_LOAD = 1, CI_IDENT = 2 };

// Wave-level tiled f32 GEMM: D = (NEGA? -A : A)[T?] x B[T?] + Cinit
// 16x16 tiles, K stepped by 4 via V_WMMA_F32_16X16X4_F32 (K compile-time).
// A fragments are double-buffered in chunks of KC k-steps so a batch of DS
// loads is in flight while the previous chunk's wmmas drain (instead of the
// 1-deep load->wait->wmma chain the scheduler produced in round 1).
template<int TA, int TB, int NEGA, int CI, int K>
__device__ __forceinline__ void wave_mm(
    const float* __restrict__ A, int lda,
    const float* __restrict__ Bm, int ldb,
    const float* __restrict__ C0, int ldc,
    float* __restrict__ D, int ldd,
    int M, int N, int wave, int nwaves, int lane)
{
  constexpr int STEPS = K / 4;
  constexpr int KC = (STEPS < 8) ? STEPS : 8;
  constexpr int NC = STEPS / KC;
  const int rc = lane & 15;
  const int hi = lane >> 4;
  const int nt = N >> 4;
  const int tiles = (M >> 4) * nt;
  for (int t = wave; t < tiles; t += nwaves) {
    const int m0 = (t / nt) << 4;
    const int n0 = (t % nt) << 4;
    v8f c;
#pragma unroll
    for (int v = 0; v < 8; ++v) {
      if (CI == CI_ZERO)      c[v] = 0.0f;
      else if (CI == CI_LOAD) c[v] = C0[(m0 + v + 8 * hi) * ldc + n0 + rc];
      else                    c[v] = (m0 + v + 8 * hi == n0 + rc) ? 1.0f : 0.0f;
    }
    v2f abuf[2][KC];
#pragma unroll
    for (int j = 0; j < KC; ++j) {   // prologue: chunk 0
      const int k0 = 4 * j;
      if (TA) {
        abuf[0][j].x = A[(k0 + 2 * hi) * lda + m0 + rc];
        abuf[0][j].y = A[(k0 + 2 * hi + 1) * lda + m0 + rc];
      } else {
        abuf[0][j].x = A[(m0 + rc) * lda + k0 + 2 * hi];
        abuf[0][j].y = A[(m0 + rc) * lda + k0 + 2 * hi + 1];
      }
    }
#pragma unroll
    for (int ch = 0; ch < NC; ++ch) {
      if (ch + 1 < NC) {             // prefetch next chunk
#pragma unroll
        for (int j = 0; j < KC; ++j) {
          const int k0 = 4 * ((ch + 1) * KC + j);
          v2f a;
          if (TA) {
            a.x = A[(k0 + 2 * hi) * lda + m0 + rc];
            a.y = A[(k0 + 2 * hi + 1) * lda + m0 + rc];
          } else {
            a.x = A[(m0 + rc) * lda + k0 + 2 * hi];
            a.y = A[(m0 + rc) * lda + k0 + 2 * hi + 1];
          }
          abuf[(ch + 1) & 1][j] = a;
        }
      }
#pragma unroll
      for (int j = 0; j < KC; ++j) {
        const int k0 = 4 * (ch * KC + j);
        v2f a = abuf[ch & 1][j];
        if (NEGA) { a.x = -a.x; a.y = -a.y; }
        v2f b;
        if (TB) {
          b.x = Bm[(n0 + rc) * ldb + k0 + 2 * hi];
          b.y = Bm[(n0 + rc) * ldb + k0 + 2 * hi + 1];
        } else {
          b.x = Bm[(k0 + 2 * hi) * ldb + n0 + rc];
          b.y = Bm[(k0 + 2 * hi + 1) * ldb + n0 + rc];
        }
        c = __builtin_amdgcn_wmma_f32_16x16x4_f32(false, a, false, b, (short)0,
                                                  c, false, false);
      }
    }
#pragma unroll
    for (int v = 0; v < 8; ++v)
      D[(m0 + v + 8 * hi) * ldd + n0 + rc] = c[v];
  }
}

// ---------------- Phase 1: sequential P/L recursion, 1 block ----------------
// Writes L_used[t]^T ([NY][NX] row-major, unpadded) for t = 0..T-1 into LT.
__global__ void __launch_bounds__(256) kf_gain_kernel(
    const float* __restrict__ Wfx, const float* __restrict__ Wfy,
    const float* __restrict__ Q, const float* __restrict__ R,
    const float* __restrict__ P0, const float* __restrict__ L0,
    float* __restrict__ LT)
{
  extern __shared__ float smem[];
  float* sP    = smem;                 // NX x PX
  float* sT    = sP + NX * PX;         // NX x PX
  float* sWfxT = sT + NX * PX;         // Wfx^T, NX x PX (WfxT[k][m] = Wfx[m][k])
  float* sPH   = sWfxT + NX * PX;      // NX x PY  (reused as WtP: NY x PX)
  float* sSaug = sPH + NX * PY;        // NY x PX  augmented [S | I]
  float* sL    = sSaug + NY * PX;      // NX x PY

  const int tid = threadIdx.x;
  const int wave = tid >> 5, lane = tid & 31;

  for (int i = tid; i < NX * NX; i += 256) {
    int r = i >> 7, cidx = i & (NX - 1);
    sP[r * PX + cidx] = P0[i];
    sWfxT[cidx * PX + r] = Wfx[i];
  }
  for (int i = tid; i < NX * NY; i += 256) { // LT[0] = L0^T
    int k = i >> 7, n = i & (NX - 1);
    LT[i] = L0[n * NY + k];
  }
  __syncthreads();

  for (int t = 0; t < T_STEPS; ++t) {
    // T1 = P @ Wfx^T   (B = WfxT in LDS, row-major)
    wave_mm<0, 0, 0, CI_ZERO, NX>(sP, PX, sWfxT, PX, nullptr, 0, sT, PX,
                                  NX, NX, wave, 8, lane);
    __syncthreads();
    // P = Wfx @ T1 + Q   (A = Wfx via transposed read of WfxT; lane-contiguous)
    wave_mm<1, 0, 0, CI_LOAD, NX>(sWfxT, PX, sT, PX, Q, NX, sP, PX,
                                  NX, NX, wave, 8, lane);
    __syncthreads();
    // PH = P @ Wfy
    wave_mm<0, 0, 0, CI_ZERO, NX>(sP, PX, Wfy, NY, nullptr, 0, sPH, PY,
                                  NX, NY, wave, 8, lane);
    __syncthreads();
    // Saug[:, :64] = Wfy^T @ PH + R ;  Saug[:, 64:] = I
    wave_mm<1, 0, 0, CI_LOAD, NX>(Wfy, NY, sPH, PY, R, NY, sSaug, PX,
                                  NY, NY, wave, 8, lane);
    for (int i = tid; i < NY * NY; i += 256) {
      int r = i >> 6, cidx = i & (NY - 1);
      sSaug[r * PX + 64 + cidx] = (r == cidx) ? 1.0f : 0.0f;
    }
    __syncthreads();
    // Gauss-Jordan inverse of S (no pivoting; S = R + H^T P H is SPD-ish)
    {
      const int r0 = tid >> 7;       // 0 or 1: two rows per pass
      const int cc = tid & 127;
      for (int j = 0; j < NY; ++j) {
        float piv = sSaug[j * PX + j];
        __syncthreads();
        float pinv = 1.0f / piv;
        if (tid < 128) sSaug[j * PX + tid] *= pinv;
        __syncthreads();
        float fs[32];
#pragma unroll
        for (int it = 0; it < 32; ++it) fs[it] = sSaug[(r0 + 2 * it) * PX + j];
        __syncthreads();
#pragma unroll
        for (int it = 0; it < 32; ++it) {
          int i = r0 + 2 * it;
          if (i != j) sSaug[i * PX + cc] -= fs[it] * sSaug[j * PX + cc];
        }
        __syncthreads();
      }
    }
    // L = PH @ Sinv
    wave_mm<0, 0, 0, CI_ZERO, NY>(sPH, PY, sSaug + 64, PX, nullptr, 0, sL, PY,
                                  NX, NY, wave, 8, lane);
    __syncthreads();
    // store L^T for the NEXT step's correction, and WtP = Wfy^T @ P
    if (t + 1 < T_STEPS) {
      for (int i = tid; i < NX * NY; i += 256) {
        int k = i >> 7, n = i & (NX - 1);
        LT[(size_t)(t + 1) * NX * NY + i] = sL[n * PY + k];
      }
    }
    wave_mm<1, 0, 0, CI_ZERO, NX>(Wfy, NY, sP, PX, nullptr, 0, sPH, PX,
                                  NY, NX, wave, 8, lane);
    __syncthreads();
    // P = I - L @ WtP   (A negated, C = identity)
    wave_mm<0, 0, 1, CI_IDENT, NY>(sL, PY, sPH, PX, nullptr, 0, sP, PX,
                                   NX, NX, wave, 8, lane);
    __syncthreads();
  }
}

// -------- Phase 2: batch state recursion, 16 blocks x 16 batch rows --------
__global__ void __launch_bounds__(256) kf_state_kernel(
    const float* __restrict__ Yp, const float* __restrict__ Up,
    const float* __restrict__ Dp,
    const float* __restrict__ Wfx, const float* __restrict__ bfx,
    const float* __restrict__ Wfu, const float* __restrict__ bfu,
    const float* __restrict__ Wfd, const float* __restrict__ bfd,
    const float* __restrict__ Wfy, const float* __restrict__ bfy,
    const float* __restrict__ x0,
    const float* __restrict__ LT, float* __restrict__ out)
{
  extern __shared__ float smem[];
  float* sWfx  = smem;                 // NX x PX
  float* sWfy  = sWfx + NX * PX;       // NX x PY
  float* sWfu  = sWfy + NX * PY;       // NU x PX
  float* sWfd  = sWfu + NU * PX;       // ND x PX
  float* sBsum = sWfd + ND * PX;       // NX (bfx+bfu+bfd)
  float* sBfy  = sBsum + NX;           // NY
  float* sX    = sBfy + NY;            // ROWS x PX current state
  float* sXp   = sX + ROWS * PX;       // ROWS x PX predicted state
  float* sYm   = sXp + ROWS * PX;      // ROWS x PY  (ym - bfy)
  float* sU    = sYm + ROWS * PY;      // ROWS x PU
  float* sD    = sU + ROWS * PU;       // ROWS x PD
  float* sInn  = sD + ROWS * PD;       // ROWS x PY innovation
  float* sLT   = sInn + ROWS * PY;     // NY x PX  L_t^T

  const int tid = threadIdx.x;
  const int wave = tid >> 5, lane = tid & 31;
  const int r0 = blockIdx.x * ROWS;

  for (int i = tid; i < NX * NX; i += 256)
    sWfx[(i >> 7) * PX + (i & (NX - 1))] = Wfx[i];
  for (int i = tid; i < NX * NY; i += 256)
    sWfy[(i >> 6) * PY + (i & (NY - 1))] = Wfy[i];
  for (int i = tid; i < NU * NX; i += 256)
    sWfu[(i >> 7) * PX + (i & (NX - 1))] = Wfu[i];
  for (int i = tid; i < ND * NX; i += 256)
    sWfd[(i >> 7) * PX + (i & (NX - 1))] = Wfd[i];
  if (tid < NX) sBsum[tid] = bfx[tid] + bfu[tid] + bfd[tid];
  if (tid < NY) sBfy[tid] = bfy[tid];
  for (int i = tid; i < ROWS * NX; i += 256)
    sX[(i >> 7) * PX + (i & (NX - 1))] = x0[i & (NX - 1)];
  __syncthreads();

  for (int t = 0; t < T_STEPS; ++t) {
    const float* gY = Yp + ((size_t)t * BATCH + r0) * NY;
    const float* gU = Up + ((size_t)t * BATCH + r0) * NU;
    const float* gD = Dp + ((size_t)t * BATCH + r0) * ND;
    const float* gL = LT + (size_t)t * NX * NY;
    for (int i = tid; i < ROWS * NY; i += 256)
      sYm[(i >> 6) * PY + (i & (NY - 1))] = gY[i] - sBfy[i & (NY - 1)];
    for (int i = tid; i < ROWS * NU; i += 256)
      sU[(i >> 5) * PU + (i & (NU - 1))] = gU[i];
    for (int i = tid; i < ROWS * ND; i += 256)
      sD[(i >> 4) * PD + (i & (ND - 1))] = gD[i];
    for (int i = tid; i < NY * NX; i += 256)
      sLT[(i >> 7) * PX + (i & (NX - 1))] = gL[i];
    if (t + 1 < T_STEPS) { // hide next-step latency: global_prefetch_b8
      __builtin_prefetch(Yp + ((size_t)(t + 1) * BATCH + r0) * NY + tid * 4, 0, 1);
      __builtin_prefetch(Up + ((size_t)(t + 1) * BATCH + r0) * NU + tid * 2, 0, 1);
      __builtin_prefetch(Dp + ((size_t)(t + 1) * BATCH + r0) * ND + tid, 0, 1);
      __builtin_prefetch(LT + (size_t)(t + 1) * NX * NY + tid * 32, 0, 1);
    }
    __syncthreads();
    // xpred = x@Wfx + u@Wfu + d@Wfd + (bfx+bfu+bfd)   (bias via ldc=0 bcast)
    wave_mm<0, 0, 0, CI_LOAD, NX>(sX, PX, sWfx, PX, sBsum, 0, sXp, PX,
                                  ROWS, NX, wave, 8, lane);
    wave_mm<0, 0, 0, CI_LOAD, NU>(sU, PU, sWfu, PX, sXp, PX, sXp, PX,
                                  ROWS, NX, wave, 8, lane);
    wave_mm<0, 0, 0, CI_LOAD, ND>(sD, PD, sWfd, PX, sXp, PX, sXp, PX,
                                  ROWS, NX, wave, 8, lane);
    __syncthreads();
    // innov = (ym - bfy) - xpred@Wfy
    wave_mm<0, 0, 1, CI_LOAD, NX>(sXp, PX, sWfy, PY, sYm, PY, sInn, PY,
                                  ROWS, NY, wave, 8, lane);
    __syncthreads();
    // x = xpred + innov @ L_t^T
    wave_mm<0, 0, 0, CI_LOAD, NY>(sInn, PY, sLT, PX, sXp, PX, sX, PX,
                                  ROWS, NX, wave, 8, lane);
    __syncthreads();
  }
  for (int i = tid; i < ROWS * NX; i += 256)
    out[(size_t)r0 * NX + i] = sX[(i >> 7) * PX + (i & (NX - 1))];
}

extern "C" void kernel_launch(void* const* d_in, const int* in_sizes, int n_in,
                              void* d_out, int out_size, void* d_ws, size_t ws_size,
                              hipStream_t stream) {
  (void)in_sizes; (void)n_in; (void)out_size; (void)ws_size;
  const float* Yp  = (const float*)d_in[0];
  const float* Up  = (const float*)d_in[1];
  const float* Dp  = (const float*)d_in[2];
  const float* Wfx = (const float*)d_in[3];
  const float* bfx = (const float*)d_in[4];
  const float* Wfu = (const float*)d_in[5];
  const float* bfu = (const float*)d_in[6];
  const float* Wfd = (const float*)d_in[7];
  const float* bfd = (const float*)d_in[8];
  const float* Wfy = (const float*)d_in[9];
  const float* bfy = (const float*)d_in[10];
  const float* Q   = (const float*)d_in[11];
  const float* R   = (const float*)d_in[12];
  const float* P0  = (const float*)d_in[13];
  const float* L0  = (const float*)d_in[14];
  const float* x0  = (const float*)d_in[15];
  float* LT  = (float*)d_ws;   // T * NY * NX floats = 64 MB of L^T history
  float* out = (float*)d_out;

  const size_t sm1 = (size_t)(3 * NX * PX + NX * PY + NY * PX + NX * PY) *
                     sizeof(float);
  const size_t sm2 = (size_t)(NX * PX + NX * PY + NU * PX + ND * PX + NX + NY +
                              2 * ROWS * PX + ROWS * PY + ROWS * PU + ROWS * PD +
                              ROWS * PY + NY * PX) * sizeof(float);
  (void)hipFuncSetAttribute(reinterpret_cast<const void*>(&kf_gain_kernel),
                            hipFuncAttributeMaxDynamicSharedMemorySize, (int)sm1);
  (void)hipFuncSetAttribute(reinterpret_cast<const void*>(&kf_state_kernel),
                            hipFuncAttributeMaxDynamicSharedMemorySize, (int)sm2);

  hipLaunchKernelGGL(kf_gain_kernel, dim3(1), dim3(256), sm1, stream,
                     Wfx, Wfy, Q, R, P0, L0, LT);
  hipLaunchKernelGGL(kf_state_kernel, dim3(BATCH / ROWS), dim3(256), sm2, stream,
                     Yp, Up, Dp, Wfx, bfx, Wfu, bfu, Wfd, bfd, Wfy, bfy, x0, LT, out);
}